// SIMHardSearch_47768626266789
// MI455X (gfx1250) — compile-verified
//
#include <hip/hip_runtime.h>
#include <hip/hip_bf16.h>
#include <stdint.h>

// Problem: SIM hard search (per-column ordered stream compaction).
//   out[pos, b] = pos-th value of user_seq[:, b] whose topic == target_topic[b],
//   for pos < K, in original sequence order; zero otherwise.
//
// Tiling: block = NTHREADS threads = COLS columns x RSPLIT row-chunks.
//   thread (c, r): column b = blockIdx.x*COLS + c, rows [r*chunk, (r+1)*chunk).
//   A wave32 = 32 consecutive c -> coalesced 128B topic loads.

#define COLS     32
#define RSPLIT   16
#define NTHREADS (COLS * RSPLIT)   // 512
#define GRP      8                 // rows staged per async group
#define MAXCHUNK 128               // L / RSPLIT for L=2048 (rowids fit in u8)

// CDNA5 async global->LDS path (ASYNCcnt). Fallback to plain loads if the
// builtin is absent on this toolchain.
#if defined(__HIP_DEVICE_COMPILE__) && __has_builtin(__builtin_amdgcn_global_load_async_to_lds_b32)
#define USE_ASYNC_LDS 1
#else
#define USE_ASYNC_LDS 0
#endif

typedef __attribute__((address_space(1))) int as1_int;   // global (device) int
typedef __attribute__((address_space(3))) int as3_int;   // LDS int

__global__ __launch_bounds__(NTHREADS)
void sim_hard_search_kernel(const float* __restrict__ user_seq,
                            const int*   __restrict__ topics,
                            const int*   __restrict__ tgt_topic,
                            float*       __restrict__ out,
                            int B, int K, int chunk) {
    __shared__ int s_counts[RSPLIT][COLS];
    __shared__ unsigned char s_rowid[NTHREADS][MAXCHUNK];
#if USE_ASYNC_LDS
    __shared__ __align__(16) int s_stage[NTHREADS][2][GRP];
#endif

    const int tid  = threadIdx.x;
    const int c    = tid & (COLS - 1);
    const int r    = tid / COLS;
    const int b    = blockIdx.x * COLS + c;
    const int tt   = tgt_topic[b];
    const int row0 = r * chunk;
    const int base = row0 * B + b;

    unsigned char* myids = s_rowid[tid];
    int n = 0;

    // ---------------- Phase A: scan own chunk, record match row ids --------
#if USE_ASYNC_LDS
    {
        const int ngroups = chunk / GRP;
        // Prologue: issue async loads for group 0 (per-lane coalesced global
        // addresses, per-lane private LDS stage slots).
        {
            const int* gs = topics + base;
            #pragma unroll
            for (int j = 0; j < GRP; ++j)
                __builtin_amdgcn_global_load_async_to_lds_b32(
                    (as1_int*)(gs + j * B), (as3_int*)&s_stage[tid][0][j], 0, 0);
        }
        for (int g = 0; g < ngroups; ++g) {
            if (g + 1 < ngroups) {
                const int* gs = topics + base + (g + 1) * GRP * B;
                #pragma unroll
                for (int j = 0; j < GRP; ++j)
                    __builtin_amdgcn_global_load_async_to_lds_b32(
                        (as1_int*)(gs + j * B),
                        (as3_int*)&s_stage[tid][(g + 1) & 1][j], 0, 0);
                // Up to 16 outstanding; async loads complete in order, so
                // <=GRP outstanding implies group g has fully landed in LDS.
#if __has_builtin(__builtin_amdgcn_s_wait_asynccnt)
                __builtin_amdgcn_s_wait_asynccnt(GRP);
#else
                asm volatile("s_wait_asynccnt %0" :: "i"(GRP));
#endif
            } else {
#if __has_builtin(__builtin_amdgcn_s_wait_asynccnt)
                __builtin_amdgcn_s_wait_asynccnt(0);
#else
                asm volatile("s_wait_asynccnt 0");
#endif
            }
            const int4 t0 = *(const int4*)&s_stage[tid][g & 1][0];
            const int4 t1 = *(const int4*)&s_stage[tid][g & 1][4];
            const int gi = g * GRP;
            if (t0.x == tt) myids[n++] = (unsigned char)(gi + 0);
            if (t0.y == tt) myids[n++] = (unsigned char)(gi + 1);
            if (t0.z == tt) myids[n++] = (unsigned char)(gi + 2);
            if (t0.w == tt) myids[n++] = (unsigned char)(gi + 3);
            if (t1.x == tt) myids[n++] = (unsigned char)(gi + 4);
            if (t1.y == tt) myids[n++] = (unsigned char)(gi + 5);
            if (t1.z == tt) myids[n++] = (unsigned char)(gi + 6);
            if (t1.w == tt) myids[n++] = (unsigned char)(gi + 7);
        }
    }
#else
    #pragma unroll 4
    for (int i = 0; i < chunk; ++i) {
        if (topics[base + i * B] == tt)
            myids[n++] = (unsigned char)i;
    }
#endif

    s_counts[r][c] = n;
    __syncthreads();   // split barrier (s_barrier_signal/-wait) on gfx1250

    // ---------------- Phase B: per-column prefix over chunks ---------------
    int prefix = 0, total = 0;
    #pragma unroll
    for (int rr = 0; rr < RSPLIT; ++rr) {
        const int cnt = s_counts[rr][c];   // lanes read consecutive words: conflict-free
        prefix += (rr < r) ? cnt : 0;
        total  += cnt;
    }

    // ---------------- Phase C: emit matched values (stable order) ----------
    for (int i = 0; i < n; ++i) {
        const int pos = prefix + i;
        if (pos < K) {
            const int row = row0 + (int)myids[i];
            out[pos * B + b] = user_seq[row * B + b];  // value loaded only on match
        }
    }

    // ---------------- Phase D: zero-fill invalid tail slots ----------------
    const int zbeg = (total < K ? total : K) + r;
    for (int s = zbeg; s < K; s += RSPLIT)
        out[s * B + b] = 0.0f;
}

extern "C" void kernel_launch(void* const* d_in, const int* in_sizes, int n_in,
                              void* d_out, int out_size, void* d_ws, size_t ws_size,
                              hipStream_t stream) {
    (void)n_in; (void)d_ws; (void)ws_size;
    // setup_inputs order: user_seq(L,B) f32 | target_item(B) f32 (unused) |
    //                     user_seq_topics(L,B) int | target_item_topic(B) int |
    //                     top_k scalar
    const float* user_seq  = (const float*)d_in[0];
    const int*   topics    = (const int*)d_in[2];
    const int*   tgt_topic = (const int*)d_in[3];
    float*       out       = (float*)d_out;

    const int B = in_sizes[1];            // 4096
    const int L = in_sizes[0] / B;        // 2048
    const int K = out_size / B;           // 256 (== top_k)
    const int chunk = L / RSPLIT;         // 128

    const dim3 grid(B / COLS);            // 128 blocks
    const dim3 block(NTHREADS);           // 512 threads = 16 wave32s
    sim_hard_search_kernel<<<grid, block, 0, stream>>>(user_seq, topics, tgt_topic,
                                                       out, B, K, chunk);
}